// RhythmNetLoss_48825188221141
// MI455X (gfx1250) — compile-verified
//
#include <hip/hip_runtime.h>
#include <hip/hip_bf16.h>

typedef float v4f __attribute__((ext_vector_type(4)));
typedef float v2f __attribute__((ext_vector_type(2)));
typedef float v8f __attribute__((ext_vector_type(8)));

#define NBLOCKS 2048
#define NTHREADS 256
#define LAMBDA_VAL 100.0f

__device__ __forceinline__ float wave_reduce_add(float v) {
#pragma unroll
    for (int off = 16; off > 0; off >>= 1)
        v += __shfl_xor(v, off, 32);
    return v;
}

// ---------------- Pass 1: per-block partials of sum|r-t| and sum(g) ----------------
__global__ __launch_bounds__(NTHREADS) void rn_pass1(
    const float* __restrict__ r, const float* __restrict__ t,
    const float* __restrict__ g, float* __restrict__ pa,
    float* __restrict__ pg, long long n) {
    const long long nvec   = n >> 2;
    const long long tid    = (long long)blockIdx.x * blockDim.x + threadIdx.x;
    const long long stride = (long long)gridDim.x * blockDim.x;
    const v4f* __restrict__ r4 = (const v4f*)r;
    const v4f* __restrict__ t4 = (const v4f*)t;
    const v4f* __restrict__ g4 = (const v4f*)g;

    float sa = 0.0f, sg = 0.0f;
    for (long long i = tid; i < nvec; i += stride) {
        // r and t are read exactly once in the whole problem: stream them (NT)
        // so they do not evict g from the 192MB L2 before pass 2.
        v4f rv = __builtin_nontemporal_load(&r4[i]);
        v4f tv = __builtin_nontemporal_load(&t4[i]);
        v4f gv = g4[i];  // keep g cache-resident (RT)
        sa += __builtin_fabsf(rv.x - tv.x) + __builtin_fabsf(rv.y - tv.y) +
              __builtin_fabsf(rv.z - tv.z) + __builtin_fabsf(rv.w - tv.w);
        sg += gv.x + gv.y + gv.z + gv.w;
    }
    if (tid == 0) { // scalar tail (n not multiple of 4)
        for (long long i = nvec << 2; i < n; ++i) {
            sa += __builtin_fabsf(r[i] - t[i]);
            sg += g[i];
        }
    }

    __shared__ float reda[NTHREADS / 32];
    __shared__ float redg[NTHREADS / 32];
    sa = wave_reduce_add(sa);
    sg = wave_reduce_add(sg);
    const int lane = threadIdx.x & 31, wid = threadIdx.x >> 5;
    if (lane == 0) { reda[wid] = sa; redg[wid] = sg; }
    __syncthreads();
    if (threadIdx.x == 0) {
        float a = 0.0f, gs = 0.0f;
#pragma unroll
        for (int w = 0; w < NTHREADS / 32; ++w) { a += reda[w]; gs += redg[w]; }
        pa[blockIdx.x] = a;
        pg[blockIdx.x] = gs;
    }
}

// WMMA-based reduction of a 64-float chunk stream: D += A(16x4) * ones(4x16).
// A layout (32-bit A 16x4): lanes 0-15 hold (M=lane, K=0,1); lanes 16-31 hold
// (M=lane-16, K=2,3). So element P[base + m*4 + k] -> lane/vgpr as below.
__device__ __forceinline__ void wmma_accum_chunk(const float* __restrict__ p,
                                                 int base, int m, int koff,
                                                 v2f ones, v8f& c) {
    v2f a;
    const int idx = base + m * 4 + koff;
    a.x = p[idx];
    a.y = p[idx + 1];
    c = __builtin_amdgcn_wmma_f32_16x16x4_f32(false, a, false, ones, (short)0, c,
                                              false, false);
}

// Extract total sum from accumulator: C/D layout -> lane l(0-15) col N=l rows M=j,
// lanes 16-31 col N=l-16 rows M=j+8. All columns identical (B = ones), so
// total = sum_j c[j](lane0) + sum_j c[j](lane16).
__device__ __forceinline__ float wmma_total(v8f c) {
    float s = 0.0f;
#pragma unroll
    for (int j = 0; j < 8; ++j) s += c[j];
    s += __shfl_xor(s, 16, 32);
    return s;
}

// ---------------- Reduce 1: 2048 partials (x2) -> l1_sum, mean_g ----------------
__global__ void rn_reduce1(const float* __restrict__ pa,
                           const float* __restrict__ pg,
                           float* __restrict__ scalars, float inv_n) {
    const int l = threadIdx.x;
    const int m = l & 15;
    const int koff = (l >> 4) << 1;
    v2f ones; ones.x = 1.0f; ones.y = 1.0f;
    v8f ca = {};
    v8f cg = {};
    for (int base = 0; base < NBLOCKS; base += 64) {
        wmma_accum_chunk(pa, base, m, koff, ones, ca);
        wmma_accum_chunk(pg, base, m, koff, ones, cg);
    }
    const float sa = wmma_total(ca);
    const float sg = wmma_total(cg);
    if (l == 0) {
        scalars[0] = sa;           // l1 sum (un-normalized)
        scalars[1] = sg * inv_n;   // mean(g)
    }
}

// ---------------- Pass 2: per-block partials of sum|g - mean_g| ----------------
__global__ __launch_bounds__(NTHREADS) void rn_pass2(
    const float* __restrict__ g, const float* __restrict__ scalars,
    float* __restrict__ ps, long long n) {
    const float mu = scalars[1];
    const long long nvec   = n >> 2;
    const long long tid    = (long long)blockIdx.x * blockDim.x + threadIdx.x;
    const long long stride = (long long)gridDim.x * blockDim.x;
    const v4f* __restrict__ g4 = (const v4f*)g;

    float ss = 0.0f;
    for (long long i = tid; i < nvec; i += stride) {
        v4f gv = g4[i];  // should hit L2 (192MB) thanks to NT hints in pass 1
        ss += __builtin_fabsf(gv.x - mu) + __builtin_fabsf(gv.y - mu) +
              __builtin_fabsf(gv.z - mu) + __builtin_fabsf(gv.w - mu);
    }
    if (tid == 0) {
        for (long long i = nvec << 2; i < n; ++i) ss += __builtin_fabsf(g[i] - mu);
    }

    __shared__ float reds[NTHREADS / 32];
    ss = wave_reduce_add(ss);
    const int lane = threadIdx.x & 31, wid = threadIdx.x >> 5;
    if (lane == 0) reds[wid] = ss;
    __syncthreads();
    if (threadIdx.x == 0) {
        float s = 0.0f;
#pragma unroll
        for (int w = 0; w < NTHREADS / 32; ++w) s += reds[w];
        ps[blockIdx.x] = s;
    }
}

// ---------------- Reduce 2: smooth partials -> final scalar ----------------
__global__ void rn_reduce2(const float* __restrict__ ps,
                           const float* __restrict__ scalars,
                           float* __restrict__ out, float inv_n) {
    const int l = threadIdx.x;
    const int m = l & 15;
    const int koff = (l >> 4) << 1;
    v2f ones; ones.x = 1.0f; ones.y = 1.0f;
    v8f cs = {};
    for (int base = 0; base < NBLOCKS; base += 64) {
        wmma_accum_chunk(ps, base, m, koff, ones, cs);
    }
    const float ssum = wmma_total(cs);
    if (l == 0) {
        out[0] = scalars[0] * inv_n + LAMBDA_VAL * (ssum * inv_n);
    }
}

extern "C" void kernel_launch(void* const* d_in, const int* in_sizes, int n_in,
                              void* d_out, int out_size, void* d_ws, size_t ws_size,
                              hipStream_t stream) {
    const float* r = (const float*)d_in[0];  // resnet_outputs
    const float* g = (const float*)d_in[1];  // gru_outputs
    const float* t = (const float*)d_in[2];  // target
    const long long n = (long long)in_sizes[0];
    const float inv_n = 1.0f / (float)n;

    float* ws      = (float*)d_ws;
    float* pa      = ws;               // [0, 2048)   : |r-t| block partials
    float* pg      = ws + NBLOCKS;     // [2048, 4096): g block partials
    float* ps      = ws + 2 * NBLOCKS; // [4096, 6144): |g-mu| block partials
    float* scalars = ws + 3 * NBLOCKS; // [6144, 6146): l1_sum, mean_g

    rn_pass1<<<NBLOCKS, NTHREADS, 0, stream>>>(r, t, g, pa, pg, n);
    rn_reduce1<<<1, 32, 0, stream>>>(pa, pg, scalars, inv_n);
    rn_pass2<<<NBLOCKS, NTHREADS, 0, stream>>>(g, scalars, ps, n);
    rn_reduce2<<<1, 32, 0, stream>>>(ps, scalars, (float*)d_out, inv_n);
}